// RGCN_68049461837951
// MI455X (gfx1250) — compile-verified
//
#include <hip/hip_runtime.h>
#include <hip/hip_bf16.h>
#include <math.h>

#define N_NODES    100000
#define N_EDGES    1600000
#define NUM_REL    4
#define NUM_GRAPHS 256

typedef __attribute__((ext_vector_type(2))) float v2f;
typedef __attribute__((ext_vector_type(8))) float v8f;

// ---------------------------------------------------------------------------
// Dense transform H[m,n] = sum_k X[m,k] * W[k,n] (+bias) using
// V_WMMA_F32_16X16X4_F32. One wave (32 lanes) computes one 16x16 output tile.
// Templated on DIN/DOUT so the K loop fully unrolls into back-to-back WMMAs.
// ---------------------------------------------------------------------------
template <int DIN, int DOUT>
__global__ __launch_bounds__(32)
void transform_wmma(const float* __restrict__ X,
                    const float* __restrict__ W,
                    const float* __restrict__ bias,   // nullptr => no bias
                    float* __restrict__ H) {
  const int lane = threadIdx.x & 31;
  const int half = lane >> 4;   // 0: lanes 0-15, 1: lanes 16-31
  const int l16  = lane & 15;
  const int m0   = blockIdx.x * 16;
  const int n0   = blockIdx.y * 16;
  const int row  = m0 + l16;

  v8f acc = {};
#pragma unroll
  for (int k0 = 0; k0 < DIN; k0 += 4) {
    // A 16x4: VGPR v holds K = k0 + v + 2*half, M = l16
    // B 4x16: VGPR v holds K = k0 + v + 2*half, N = n0 + l16
    const int ka = k0 + 2 * half;
    v2f a, b;
    a.x = X[(size_t)row * DIN + ka];
    a.y = X[(size_t)row * DIN + ka + 1];
    b.x = W[(size_t)(ka    ) * DOUT + n0 + l16];
    b.y = W[(size_t)(ka + 1) * DOUT + n0 + l16];
    acc = __builtin_amdgcn_wmma_f32_16x16x4_f32(false, a, false, b,
                                                (short)0, acc, false, false);
  }

  const int   col = n0 + l16;
  const float bv  = bias ? bias[col] : 0.0f;
  // C/D: VGPR v holds M = v + 8*half, N = l16
#pragma unroll
  for (int v = 0; v < 8; ++v) {
    H[(size_t)(m0 + v + 8 * half) * DOUT + col] = acc[v] + bv;
  }
}

// ---------------------------------------------------------------------------
// Edge-count per (dst, relation) segment
// ---------------------------------------------------------------------------
__global__ void zero_i32(int* __restrict__ p, int n) {
  int t = blockIdx.x * blockDim.x + threadIdx.x;
  if (t < n) p[t] = 0;
}

__global__ void count_edges(const int* __restrict__ dst,
                            const int* __restrict__ et,
                            int* __restrict__ cnt) {
  int e = blockIdx.x * blockDim.x + threadIdx.x;
  if (e < N_EDGES) atomicAdd(&cnt[dst[e] * NUM_REL + et[e]], 1);
}

// ---------------------------------------------------------------------------
// Mean scatter: out[dst] += h[rel][src] / max(cnt[dst,rel],1).
// One thread per (edge, 4-channel quad): b128 gather from h, 4 atomic adds,
// consecutive lanes = consecutive channels of one edge => fully coalesced.
// FULLH=true : H is [R, N, DOUT], single pass over all edges.
// FULLH=false: H is [N, DOUT] for relation `rel`, skip mismatched edges.
// ---------------------------------------------------------------------------
template <int DOUT, bool FULLH>
__global__ void scatter_mean(const float* __restrict__ H,
                             const int* __restrict__ src,
                             const int* __restrict__ dst,
                             const int* __restrict__ et,
                             const int* __restrict__ cnt,
                             int rel,
                             float* __restrict__ out) {
  constexpr int Q = DOUT / 4;                 // channel quads per edge
  const long long t = (long long)blockIdx.x * blockDim.x + threadIdx.x;
  if (t >= (long long)N_EDGES * Q) return;
  const int e = (int)(t / Q);                 // Q is a power of two -> shift
  const int c = ((int)(t & (Q - 1))) * 4;
  const int r = et[e];
  if (!FULLH && r != rel) return;
  const int   d    = dst[e];
  const float norm = 1.0f / fmaxf((float)cnt[d * NUM_REL + r], 1.0f);
  const size_t hrow =
      ((FULLH ? (size_t)r * N_NODES : (size_t)0) + (size_t)src[e]) * DOUT + c;
  const float4 hv = *(const float4*)(H + hrow);
  float* o = out + (size_t)d * DOUT + c;
  atomicAdd(o + 0, hv.x * norm);
  atomicAdd(o + 1, hv.y * norm);
  atomicAdd(o + 2, hv.z * norm);
  atomicAdd(o + 3, hv.w * norm);
}

__global__ void relu_k(float* __restrict__ p, int n) {
  int t = blockIdx.x * blockDim.x + threadIdx.x;
  if (t < n) p[t] = fmaxf(p[t], 0.0f);
}

__global__ void fill_neginf(float* __restrict__ p, int n) {
  int t = blockIdx.x * blockDim.x + threadIdx.x;
  if (t < n) p[t] = -INFINITY;
}

// ---------------------------------------------------------------------------
// segment_max over graphs; float atomic max via signed/unsigned int punning
// (valid with out initialized to -inf, correct for mixed signs).
// ---------------------------------------------------------------------------
__global__ void pool_max(const float* __restrict__ h,
                         const int* __restrict__ batch,
                         float* __restrict__ out) {
  int t = blockIdx.x * blockDim.x + threadIdx.x;
  if (t >= N_NODES * 64) return;
  const int n = t >> 6;
  const int c = t & 63;
  const float v = h[t];
  float* addr = &out[batch[n] * 64 + c];
  if (v >= 0.0f) atomicMax((int*)addr, __float_as_int(v));
  else           atomicMin((unsigned int*)addr, __float_as_uint(v));
}

// ---------------------------------------------------------------------------
// One RGCN layer: out = X@root + bias; h_r = X@W_r; mean-scatter; opt. ReLU.
// ---------------------------------------------------------------------------
template <int DIN, int DOUT>
static void run_layer(const float* X, const float* W, const float* root,
                      const float* bias, const int* src, const int* dst,
                      const int* et, const int* cnt, float* h, bool fullh,
                      float* out, bool relu, hipStream_t stream) {
  dim3 tg(N_NODES / 16, DOUT / 16);
  transform_wmma<DIN, DOUT><<<tg, 32, 0, stream>>>(X, root, bias, out);

  constexpr int Q = DOUT / 4;
  const long long tot  = (long long)N_EDGES * Q;
  const int       blks = (int)((tot + 255) / 256);

  if (fullh) {
    for (int r = 0; r < NUM_REL; ++r)
      transform_wmma<DIN, DOUT><<<tg, 32, 0, stream>>>(
          X, W + (size_t)r * DIN * DOUT, nullptr,
          h + (size_t)r * N_NODES * DOUT);
    scatter_mean<DOUT, true><<<blks, 256, 0, stream>>>(h, src, dst, et, cnt,
                                                       0, out);
  } else {
    for (int r = 0; r < NUM_REL; ++r) {
      transform_wmma<DIN, DOUT><<<tg, 32, 0, stream>>>(
          X, W + (size_t)r * DIN * DOUT, nullptr, h);
      scatter_mean<DOUT, false><<<blks, 256, 0, stream>>>(h, src, dst, et,
                                                          cnt, r, out);
    }
  }
  if (relu) {
    const int n = N_NODES * DOUT;
    relu_k<<<(n + 255) / 256, 256, 0, stream>>>(out, n);
  }
}

extern "C" void kernel_launch(void* const* d_in, const int* in_sizes, int n_in,
                              void* d_out, int out_size, void* d_ws,
                              size_t ws_size, hipStream_t stream) {
  (void)in_sizes; (void)n_in; (void)out_size;

  const float* x     = (const float*)d_in[0];
  const int*   ei    = (const int*)d_in[1];
  const int*   et    = (const int*)d_in[2];
  const int*   batch = (const int*)d_in[3];
  const float* W1 = (const float*)d_in[4];
  const float* r1 = (const float*)d_in[5];
  const float* b1 = (const float*)d_in[6];
  const float* W2 = (const float*)d_in[7];
  const float* r2 = (const float*)d_in[8];
  const float* b2 = (const float*)d_in[9];
  const float* W3 = (const float*)d_in[10];
  const float* r3 = (const float*)d_in[11];
  const float* b3 = (const float*)d_in[12];
  const int* src = ei;
  const int* dst = ei + N_EDGES;
  float* out = (float*)d_out;

  // Workspace layout (16B-aligned offsets):
  //   cnt   @ 0          : N*R int32       =   1.6 MB
  //   buf2  @ 1.6e6      : N*32 f32        =  12.8 MB   (layer-2 output)
  //   buf13 @ 14.4e6     : N*64 f32        =  25.6 MB   (layer-1 & -3 output)
  //   h     @ 40.0e6     : full: R*N*64 f32 = 102.4 MB; fallback: N*64 f32
  char*  ws    = (char*)d_ws;
  int*   cnt   = (int*)ws;
  float* buf2  = (float*)(ws + 1600000);
  float* buf13 = (float*)(ws + 14400000);
  float* h     = (float*)(ws + 40000000);

  const size_t NEED_BIG = 40000000ull + 4ull * N_NODES * 64 * 4;  // 142.4 MB
  const bool fullh = ws_size >= NEED_BIG;   // fixed per harness => deterministic

  // Per-(dst,rel) edge counts (graph fixed across layers -> compute once).
  zero_i32<<<(N_NODES * NUM_REL + 255) / 256, 256, 0, stream>>>(
      cnt, N_NODES * NUM_REL);
  count_edges<<<(N_EDGES + 255) / 256, 256, 0, stream>>>(dst, et, cnt);

  // Layer 1: 16 -> 16 (+ReLU)     X = input,  out = buf13
  run_layer<16, 16>(x,     W1, r1, b1, src, dst, et, cnt, h, fullh, buf13,
                    true, stream);
  // Layer 2: 16 -> 32 (+ReLU)     X = buf13,  out = buf2
  run_layer<16, 32>(buf13, W2, r2, b2, src, dst, et, cnt, h, fullh, buf2,
                    true, stream);
  // Layer 3: 32 -> 64 (no ReLU)   X = buf2,   out = buf13
  run_layer<32, 64>(buf2,  W3, r3, b3, src, dst, et, cnt, h, fullh, buf13,
                    false, stream);

  // Graph max-pool -> [256, 64]
  fill_neginf<<<(NUM_GRAPHS * 64 + 255) / 256, 256, 0, stream>>>(
      out, NUM_GRAPHS * 64);
  pool_max<<<(N_NODES * 64 + 255) / 256, 256, 0, stream>>>(buf13, batch, out);
}